// ODEBlock_72335839199613
// MI455X (gfx1250) — compile-verified
//
#include <hip/hip_runtime.h>
#include <math.h>

typedef __attribute__((ext_vector_type(2))) float v2f;
typedef __attribute__((ext_vector_type(4))) float v4f;
typedef __attribute__((ext_vector_type(8))) float v8f;

#define HID    64
#define NROWS  6000          // 2000 S + 2000 I + 2000 R rows
#define NTIME  100
#define STEPS  99
#define DT     0.1f
#define STR    68            // padded LDS row stride (floats); even -> b64-aligned; 4*m+k banking
#define WPB    2             // waves per block (static LDS < 64KB)
#define RPW    16            // rows per wave (one 16x64 tile)

#define WT_SZ  (HID * STR)
#define BUF_SZ (RPW * STR)

__device__ __forceinline__ void wave_lds_fence() {
  asm volatile("s_wait_dscnt 0" ::: "memory");   // wave-local LDS store->load ordering
}

__device__ __forceinline__ float fast_tanh(float x) {
#if __has_builtin(__builtin_amdgcn_tanhf)
  return __builtin_amdgcn_tanhf(x);              // CDNA5 HW transcendental (TRANS32)
#else
  return tanhf(x);
#endif
}

// One f(y) evaluation for a 16x64 tile:
//   out^T = W2^T * tanh(W1^T * arg^T + b1) + b2   (tiles in WMMA C/D layout, M=feature, N=row)
__device__ __forceinline__ void feval(const float* __restrict__ W1T,
                                      const float* __restrict__ W2T,
                                      const float* __restrict__ sb1,
                                      const float* __restrict__ sb2,
                                      const float* __restrict__ argbuf, // LDS [m*STR + phi]
                                      float* __restrict__ tbuf,         // LDS [m*STR + phi]
                                      v8f (&out)[4], int lane)
{
  const int l15    = lane & 15;
  const int koff   = (lane & 16) ? 2 : 0;   // A/B layout: lanes16-31 hold K=2,3
  const int phioff = (lane & 16) ? 8 : 0;   // C/D layout: lanes16-31 hold M=r+8

  v8f acc[4];
#pragma unroll
  for (int i = 0; i < 4; ++i) {             // bias init via LDS broadcast (keeps VGPRs < 256)
    v4f lo = *(const v4f*)&sb1[i * 16 + phioff];
    v4f hi = *(const v4f*)&sb1[i * 16 + phioff + 4];
#pragma unroll
    for (int r = 0; r < 4; ++r) { acc[i][r] = lo[r]; acc[i][r + 4] = hi[r]; }
  }

  // GEMM1: acc = W1^T * arg^T + b1   (4 feature tiles x 16 k-chunks of 16x16x4)
#pragma unroll
  for (int kc = 0; kc < 16; ++kc) {
    v2f b = *(const v2f*)&argbuf[l15 * STR + kc * 4 + koff];
#pragma unroll
    for (int i = 0; i < 4; ++i) {
      v2f a = *(const v2f*)&W1T[(i * 16 + l15) * STR + kc * 4 + koff];
      acc[i] = __builtin_amdgcn_wmma_f32_16x16x4_f32(
                 false, a, false, b, (short)0, acc[i], false, false);
    }
  }

  // tanh (HW TRANS), re-stage through LDS as B operand for GEMM2 (b128 stores)
#pragma unroll
  for (int i = 0; i < 4; ++i) {
    v4f lo, hi;
#pragma unroll
    for (int r = 0; r < 4; ++r) {
      lo[r] = fast_tanh(acc[i][r]);
      hi[r] = fast_tanh(acc[i][r + 4]);
    }
    *(v4f*)&tbuf[l15 * STR + i * 16 + phioff]     = lo;
    *(v4f*)&tbuf[l15 * STR + i * 16 + phioff + 4] = hi;
  }
  wave_lds_fence();

#pragma unroll
  for (int i = 0; i < 4; ++i) {
    v4f lo = *(const v4f*)&sb2[i * 16 + phioff];
    v4f hi = *(const v4f*)&sb2[i * 16 + phioff + 4];
#pragma unroll
    for (int r = 0; r < 4; ++r) { out[i][r] = lo[r]; out[i][r + 4] = hi[r]; }
  }

  // GEMM2: out = W2^T * tanh(...)^T + b2
#pragma unroll
  for (int kc = 0; kc < 16; ++kc) {
    v2f b = *(const v2f*)&tbuf[l15 * STR + kc * 4 + koff];
#pragma unroll
    for (int i = 0; i < 4; ++i) {
      v2f a = *(const v2f*)&W2T[(i * 16 + l15) * STR + kc * 4 + koff];
      out[i] = __builtin_amdgcn_wmma_f32_16x16x4_f32(
                 false, a, false, b, (short)0, out[i], false, false);
    }
  }
}

// Store a C/D-layout tile set to the row-major LDS buffer (two b128 per tile)
__device__ __forceinline__ void store_cd(float* __restrict__ buf, const v8f (&x)[4],
                                         int l15, int phioff) {
#pragma unroll
  for (int i = 0; i < 4; ++i) {
    v4f lo, hi;
#pragma unroll
    for (int r = 0; r < 4; ++r) { lo[r] = x[i][r]; hi[r] = x[i][r + 4]; }
    *(v4f*)&buf[l15 * STR + i * 16 + phioff]     = lo;
    *(v4f*)&buf[l15 * STR + i * 16 + phioff + 4] = hi;
  }
}

__global__ void __launch_bounds__(WPB * 32)
odeblock_kernel(const float* __restrict__ S0, const float* __restrict__ I0,
                const float* __restrict__ R0, const float* __restrict__ wS1,
                const float* __restrict__ bS1, const float* __restrict__ W3,
                const float* __restrict__ b3, const float* __restrict__ WS2,
                const float* __restrict__ bS2, const float* __restrict__ Wf1,
                const float* __restrict__ bf1, const float* __restrict__ Wf2,
                const float* __restrict__ bf2, float* __restrict__ out)
{
  __shared__ float sW1T[WT_SZ];
  __shared__ float sW2T[WT_SZ];
  __shared__ float sb1[HID], sb2[HID];
  __shared__ float sW3[HID * 4];
  __shared__ float sb3[4], sWS2[4], sbS2[1];
  __shared__ float sbuf[WPB * 2 * BUF_SZ];

  const int tid = threadIdx.x;

  for (int idx = tid; idx < HID * HID; idx += blockDim.x) {
    int k = idx >> 6, phi = idx & 63;          // Wf row-major [k][phi]
    sW1T[phi * STR + k] = Wf1[idx];
    sW2T[phi * STR + k] = Wf2[idx];
  }
  for (int idx = tid; idx < HID; idx += blockDim.x) { sb1[idx] = bf1[idx]; sb2[idx] = bf2[idx]; }
  for (int idx = tid; idx < HID * 4; idx += blockDim.x) sW3[idx] = W3[idx];
  if (tid < 4) { sb3[tid] = b3[tid]; sWS2[tid] = WS2[tid]; }
  if (tid == 0) sbS2[0] = bS2[0];
  __syncthreads();

  const int wave = tid >> 5, lane = tid & 31;
  const int rowBase = (blockIdx.x * WPB + wave) * RPW;
  float* argbuf = &sbuf[wave * 2 * BUF_SZ];
  float* tbuf   = argbuf + BUF_SZ;

  const int l15    = lane & 15;
  const int phioff = (lane & 16) ? 8 : 0;

  // y0: S rows = relu(S0*wS1+bS1), I rows likewise, R rows = R0
  for (int idx = lane; idx < RPW * HID; idx += 32) {
    int m = idx >> 6, k = idx & 63;
    int g  = rowBase + m;
    int gc = g < NROWS ? g : NROWS - 1;        // clamp pad rows to valid memory
    float v;
    if (gc < 2000)      v = fmaxf(S0[gc] * wS1[k] + bS1[k], 0.0f);
    else if (gc < 4000) v = fmaxf(I0[gc - 2000] * wS1[k] + bS1[k], 0.0f);
    else                v = R0[(gc - 4000) * HID + k];
    argbuf[m * STR + k] = v;
  }
  wave_lds_fence();

  // y into C/D registers (invariant: argbuf holds current y at loop top)
  v8f y[4];
#pragma unroll
  for (int i = 0; i < 4; ++i) {
    v4f lo = *(const v4f*)&argbuf[l15 * STR + i * 16 + phioff];
    v4f hi = *(const v4f*)&argbuf[l15 * STR + i * 16 + phioff + 4];
#pragma unroll
    for (int r = 0; r < 4; ++r) { y[i][r] = lo[r]; y[i][r + 4] = hi[r]; }
  }

  const float h = DT;
  const float third = 1.0f / 3.0f;
  v8f k1[4], k2[4], k3[4], k4[4], tmp[4];

  for (int t = 0; t < NTIME; ++t) {
    // head on current y (argbuf): relu(y@W3+b3)@WS2+bS2, lanes 0-15 one row each
    if (lane < 16) {
      int g = rowBase + lane;
      if (g < NROWS) {
        float z0 = 0.f, z1 = 0.f, z2 = 0.f, z3 = 0.f;
        for (int k = 0; k < HID; ++k) {
          float yv = argbuf[lane * STR + k];
          z0 += yv * sW3[k * 4 + 0];
          z1 += yv * sW3[k * 4 + 1];
          z2 += yv * sW3[k * 4 + 2];
          z3 += yv * sW3[k * 4 + 3];
        }
        float s = sbS2[0];
        s += fmaxf(z0 + sb3[0], 0.f) * sWS2[0];
        s += fmaxf(z1 + sb3[1], 0.f) * sWS2[1];
        s += fmaxf(z2 + sb3[2], 0.f) * sWS2[2];
        s += fmaxf(z3 + sb3[3], 0.f) * sWS2[3];
        out[(g / 2000) * (NTIME * 2000) + t * 2000 + (g % 2000)] = s;  // raw score slot
      }
    }
    if (t == STEPS) break;   // final y only needs the head

    feval(sW1T, sW2T, sb1, sb2, argbuf, tbuf, k1, lane);     // k1 = f(y)

#pragma unroll
    for (int i = 0; i < 4; ++i) tmp[i] = y[i] + k1[i] * (h * third);
    store_cd(argbuf, tmp, l15, phioff);
    wave_lds_fence();
    feval(sW1T, sW2T, sb1, sb2, argbuf, tbuf, k2, lane);     // k2 = f(y + h*k1/3)

#pragma unroll
    for (int i = 0; i < 4; ++i) tmp[i] = y[i] + (k2[i] - k1[i] * third) * h;
    store_cd(argbuf, tmp, l15, phioff);
    wave_lds_fence();
    feval(sW1T, sW2T, sb1, sb2, argbuf, tbuf, k3, lane);     // k3 = f(y + h*(k2-k1/3))

#pragma unroll
    for (int i = 0; i < 4; ++i) tmp[i] = y[i] + (k1[i] - k2[i] + k3[i]) * h;
    store_cd(argbuf, tmp, l15, phioff);
    wave_lds_fence();
    feval(sW1T, sW2T, sb1, sb2, argbuf, tbuf, k4, lane);     // k4 = f(y + h*(k1-k2+k3))

#pragma unroll
    for (int i = 0; i < 4; ++i)
      y[i] = y[i] + (k1[i] + (k2[i] + k3[i]) * 3.0f + k4[i]) * (h * 0.125f);
    store_cd(argbuf, y, l15, phioff);                        // restore invariant
    wave_lds_fence();
  }
}

// In-place 3-way softmax over (S,I,R) per (t,node); d_out = [S|I|R] each NTIME*2000
__global__ void softmax3_kernel(float* __restrict__ out) {
  int i = blockIdx.x * blockDim.x + threadIdx.x;
  const int P = NTIME * 2000;
  if (i >= P) return;
  float a = out[i], b = out[P + i], c = out[2 * P + i];
  float m  = fmaxf(a, fmaxf(b, c));
  float ea = expf(a - m), eb = expf(b - m), ec = expf(c - m);
  float inv = 1.0f / (ea + eb + ec);
  out[i] = ea * inv;
  out[P + i] = eb * inv;
  out[2 * P + i] = ec * inv;
}

extern "C" void kernel_launch(void* const* d_in, const int* in_sizes, int n_in,
                              void* d_out, int out_size, void* d_ws, size_t ws_size,
                              hipStream_t stream) {
  (void)in_sizes; (void)n_in; (void)out_size; (void)d_ws; (void)ws_size;
  const float* S0  = (const float*)d_in[0];
  const float* I0  = (const float*)d_in[1];
  const float* R0  = (const float*)d_in[2];
  const float* wS1 = (const float*)d_in[3];
  const float* bS1 = (const float*)d_in[4];
  const float* W3  = (const float*)d_in[5];
  const float* b3  = (const float*)d_in[6];
  const float* WS2 = (const float*)d_in[7];
  const float* bS2 = (const float*)d_in[8];
  const float* Wf1 = (const float*)d_in[9];
  const float* bf1 = (const float*)d_in[10];
  const float* Wf2 = (const float*)d_in[11];
  const float* bf2 = (const float*)d_in[12];
  float* out = (float*)d_out;

  const int totalWaves = NROWS / RPW;                     // 375
  const int blocks = (totalWaves + WPB - 1) / WPB;        // 188 (last wave padded+guarded)
  odeblock_kernel<<<blocks, WPB * 32, 0, stream>>>(
      S0, I0, R0, wS1, bS1, W3, b3, WS2, bS2, Wf1, bf1, Wf2, bf2, out);

  const int P = NTIME * 2000;
  softmax3_kernel<<<(P + 255) / 256, 256, 0, stream>>>(out);
}